// QLSTM_65481071402450
// MI455X (gfx1250) — compile-verified
//
#include <hip/hip_runtime.h>

#define T_ 512
#define B_ 256
#define D_ 256
#define H_ 32
#define DH 288  // D + H

typedef __attribute__((ext_vector_type(2))) float v2f;
typedef __attribute__((ext_vector_type(8))) float v8f;

// ---------------------------------------------------------------------------
// Phase 1: time-invariant projection.
// G4[(t*B + b)*32 + h] = float4(pre_f, pre_i, pre_u, pre_o)
//   where pre_g = bias_g[h] + sum_{d<256} x[t,b,d] * Wg[h,d]
// GEMM: M = T*B = 131072, N = 128 (4 gates x 32), K = 256, f32 WMMA 16x16x4.
// Each wave owns one 16-row M-tile and all 8 N-tiles (so it holds every gate
// for its rows -> fused interleaved float4 store). B-matrix is staged in LDS
// pre-swizzled into per-lane v2f exactly matching the WMMA B layout
// (lane = N%16, K pair = (lane/16)*2), in 4 chunks of K=64 (32KB LDS).
// ---------------------------------------------------------------------------
__global__ __launch_bounds__(256)
void qlstm_xproj(const float* __restrict__ X,
                 const float* __restrict__ Wf, const float* __restrict__ Wi,
                 const float* __restrict__ Wu, const float* __restrict__ Wo,
                 const float* __restrict__ bf, const float* __restrict__ bi,
                 const float* __restrict__ bu, const float* __restrict__ bo,
                 float4* __restrict__ G4)
{
    __shared__ v2f Bs[16 * 8 * 32];  // [s'][nt][lane], 32 KB

    const int tid  = threadIdx.x;
    const int wave = tid >> 5;
    const int l    = tid & 31;
    const int lhi  = l >> 4;   // which K pair / which row half
    const int llo  = l & 15;

    const float* Wsel[4] = {Wf, Wi, Wu, Wo};

    const int m0 = (blockIdx.x * 8 + wave) * 16;
    // A layout (16x4 f32): lane l holds row M = l%16, K = k0 + (l/16)*2 (+0,+1)
    const float* Ap = X + (long)(m0 + llo) * D_ + (lhi << 1);

    v8f acc[8];
#pragma unroll
    for (int nt = 0; nt < 8; ++nt) {
#pragma unroll
        for (int e = 0; e < 8; ++e) acc[nt][e] = 0.0f;
    }

    for (int kc = 0; kc < 4; ++kc) {
        __syncthreads();
        // Fill this K-chunk of the swizzled B tile: 4096 v2f / 256 threads.
        for (int e = tid; e < 16 * 8 * 32; e += 256) {
            const int sp  = e >> 8;         // k-step within chunk (K = 4*sp)
            const int rem = e & 255;
            const int nt  = rem >> 5;       // N tile 0..7
            const int ll  = rem & 31;       // lane
            const int k   = (kc << 6) + (sp << 2) + ((ll >> 4) << 1);
            const int g   = nt >> 1;        // gate
            const int h   = ((nt & 1) << 4) | (ll & 15);
            const float* w = Wsel[g] + h * DH + k;  // Bmat[k][n] = Wg[h*288 + k]
            v2f v; v.x = w[0]; v.y = w[1];
            Bs[e] = v;
        }
        __syncthreads();

#pragma unroll 4
        for (int sp = 0; sp < 16; ++sp) {
            v2f a = *(const v2f*)(Ap + (kc << 6) + (sp << 2));
#pragma unroll
            for (int nt = 0; nt < 8; ++nt) {
                v2f b = Bs[(sp * 8 + nt) * 32 + l];
                acc[nt] = __builtin_amdgcn_wmma_f32_16x16x4_f32(
                    false, a, false, b, (short)0, acc[nt], false, false);
            }
        }
    }

    // C layout: VGPR r -> row M = r (lanes 0-15) / r+8 (lanes 16-31), col = l%16.
    // nt = 2*g + hb, so lane l / vgpr r holds all 4 gates for (row, h = hb*16 + l%16).
    const int rbase = m0 + (lhi << 3);
    const float4 b0 = make_float4(bf[llo],      bi[llo],      bu[llo],      bo[llo]);
    const float4 b1 = make_float4(bf[llo + 16], bi[llo + 16], bu[llo + 16], bo[llo + 16]);
#pragma unroll
    for (int r = 0; r < 8; ++r) {
        const long row = rbase + r;
        float4 v0, v1;
        v0.x = acc[0][r] + b0.x; v0.y = acc[2][r] + b0.y;
        v0.z = acc[4][r] + b0.z; v0.w = acc[6][r] + b0.w;
        v1.x = acc[1][r] + b1.x; v1.y = acc[3][r] + b1.y;
        v1.z = acc[5][r] + b1.z; v1.w = acc[7][r] + b1.w;
        G4[row * 32 + llo]      = v0;
        G4[row * 32 + 16 + llo] = v1;
    }
}

// ---------------------------------------------------------------------------
// Phase 2: the recurrence. One wave32 per batch element (B=256 independent
// sequences -> no inter-workgroup sync at all). lane = h in [0,32).
// Per step: float4 b128 load of all 4 gate preacts (L2-resident, prefetched
// one step ahead), 32x32 matvec per gate via readlane-broadcast hx with
// recurrent weights held in 128 VGPRs, qlayer cumprod as a 5-step wave32
// prefix product, then the LSTM cell update.
// ---------------------------------------------------------------------------
__device__ __forceinline__ float sigf(float x) {
    return 1.0f / (1.0f + __expf(-x));
}
__device__ __forceinline__ float tanh_fast(float x) {
    float xc = fminf(fmaxf(x, -10.0f), 10.0f);
    float e  = __expf(2.0f * xc);
    return (e - 1.0f) / (e + 1.0f);
}
__device__ __forceinline__ float qscan(float x, float ct, float ss, int lane) {
    // e = cos(x)*cos(theta) - sin(x)*(sin(phi)*sin(theta)); z = cumprod(e)
    float p = __cosf(x) * ct - __sinf(x) * ss;
#pragma unroll
    for (int off = 1; off < 32; off <<= 1) {
        float up = __shfl_up(p, off, 32);
        p = (lane >= off) ? p * up : p;
    }
    return (p + 1.0f) * 0.5f;
}

__global__ __launch_bounds__(32)
void qlstm_rec(const float4* __restrict__ G4,
               const float* __restrict__ Wf, const float* __restrict__ Wi,
               const float* __restrict__ Wu, const float* __restrict__ Wo,
               const float* __restrict__ Pf, const float* __restrict__ Pi,
               const float* __restrict__ Pu, const float* __restrict__ Po,
               float* __restrict__ out)
{
    const int b = blockIdx.x;
    const int l = threadIdx.x;  // = h

    // Recurrent weights, kept entirely in VGPRs: w[k].g = Wg[h, 256+k]
    float4 w[32];
#pragma unroll
    for (int k = 0; k < 32; ++k) {
        const int off = l * DH + D_ + k;
        w[k].x = Wf[off]; w[k].y = Wi[off]; w[k].z = Wu[off]; w[k].w = Wo[off];
    }

    // qlayer constants per (gate, h): ct = cos(theta), ss = sin(phi)*sin(theta)
    const float ctf = __cosf(Pf[l * 3 + 1]), ssf = __sinf(Pf[l * 3]) * __sinf(Pf[l * 3 + 1]);
    const float cti = __cosf(Pi[l * 3 + 1]), ssi = __sinf(Pi[l * 3]) * __sinf(Pi[l * 3 + 1]);
    const float ctu = __cosf(Pu[l * 3 + 1]), ssu = __sinf(Pu[l * 3]) * __sinf(Pu[l * 3 + 1]);
    const float cto = __cosf(Po[l * 3 + 1]), sso = __sinf(Po[l * 3]) * __sinf(Po[l * 3 + 1]);

    float hx = 0.0f, cx = 0.0f;
    float4 nxt = G4[((long)0 * B_ + b) * 32 + l];

    for (int t = 0; t < T_; ++t) {
        float4 a = nxt;
        const int tn = (t + 1 < T_) ? (t + 1) : t;      // software prefetch
        nxt = G4[((long)tn * B_ + b) * 32 + l];

#pragma unroll
        for (int k = 0; k < 32; ++k) {
            const float hk = __shfl(hx, k, 32);
            a.x += hk * w[k].x; a.y += hk * w[k].y;
            a.z += hk * w[k].z; a.w += hk * w[k].w;
        }

        const float f = sigf(qscan(a.x, ctf, ssf, l));
        const float i = sigf(qscan(a.y, cti, ssi, l));
        const float g = tanh_fast(qscan(a.z, ctu, ssu, l));
        const float o = sigf(qscan(a.w, cto, sso, l));

        cx = f * cx + i * g;
        hx = o * tanh_fast(cx);

        out[((long)t * B_ + b) * 32 + l] = hx;
    }
    out[(long)T_ * B_ * 32 + b * 32 + l]            = hx;
    out[(long)T_ * B_ * 32 + B_ * 32 + b * 32 + l]  = cx;
}

// ---------------------------------------------------------------------------
extern "C" void kernel_launch(void* const* d_in, const int* in_sizes, int n_in,
                              void* d_out, int out_size, void* d_ws, size_t ws_size,
                              hipStream_t stream) {
    (void)in_sizes; (void)n_in; (void)out_size; (void)ws_size;
    const float* X  = (const float*)d_in[0];
    const float* Wf = (const float*)d_in[1];
    const float* bf = (const float*)d_in[2];
    const float* Pf = (const float*)d_in[3];
    const float* Wi = (const float*)d_in[4];
    const float* bi = (const float*)d_in[5];
    const float* Pi = (const float*)d_in[6];
    const float* Wu = (const float*)d_in[7];
    const float* bu = (const float*)d_in[8];
    const float* Pu = (const float*)d_in[9];
    const float* Wo = (const float*)d_in[10];
    const float* bo = (const float*)d_in[11];
    const float* Po = (const float*)d_in[12];

    float4* G4 = (float4*)d_ws;        // 67 MB: T*B*32 float4 (fits in 192MB L2)
    float*  O  = (float*)d_out;

    // 8192 M-tiles of 16 rows, 8 waves per WG -> 1024 workgroups.
    qlstm_xproj<<<1024, 256, 0, stream>>>(X, Wf, Wi, Wu, Wo, bf, bi, bu, bo, G4);
    // One wave32 per batch element.
    qlstm_rec<<<256, 32, 0, stream>>>(G4, Wf, Wi, Wu, Wo, Pf, Pi, Pu, Po, O);
}